// yolov3_loss_2740189134983
// MI455X (gfx1250) — compile-verified
//
#include <hip/hip_runtime.h>
#include <cstdint>
#include <cstddef>

#define NUM_CLASSES 80
#define CCH 85          // 5 + NUM_CLASSES
#define MAX_GT 20

typedef int v4i __attribute__((ext_vector_type(4)));
typedef int v8i __attribute__((ext_vector_type(8)));

// ANCH_REV = ANCHORS[::-1]: scale 0 (13x13) uses ANCHORS[2], etc.
__constant__ float d_anch[3][3][2] = {
    {{0.2788f, 0.2163f}, {0.375f,  0.476f},  {0.8966f, 0.7837f}},
    {{0.0721f, 0.1466f}, {0.149f,  0.1082f}, {0.1418f, 0.2861f}},
    {{0.024f,  0.0313f}, {0.0385f, 0.0721f}, {0.0793f, 0.0553f}},
};

__device__ __forceinline__ float sigm(float x) { return 1.0f / (1.0f + expf(-x)); }
__device__ __forceinline__ float clipp(float p) {
    return fminf(fmaxf(p, 1e-7f), 1.0f - 1e-7f);
}
// iou(b1=gt, b2=pred), union clipped at 1e-6 like the reference
__device__ __forceinline__ float iou_one(float px, float py, float pw, float ph,
                                         float gx, float gy, float gw, float gh) {
    float tlx = fmaxf(gx - gw * 0.5f, px - pw * 0.5f);
    float tly = fmaxf(gy - gh * 0.5f, py - ph * 0.5f);
    float brx = fminf(gx + gw * 0.5f, px + pw * 0.5f);
    float bry = fminf(gy + gh * 0.5f, py + ph * 0.5f);
    float wx = fmaxf(brx - tlx, 0.0f);
    float wy = fmaxf(bry - tly, 0.0f);
    float inter = wx * wy;
    float uni = gw * gh + pw * ph - inter;
    return inter / fmaxf(uni, 1e-6f);
}

__device__ __forceinline__ int rfl(unsigned v) {
    return __builtin_amdgcn_readfirstlane((int)v);
}

template <int NT>
__global__ __launch_bounds__(NT)
void yolo_scale_loss(const float* __restrict__ pred,
                     const float* __restrict__ tgt,
                     float* __restrict__ partial,
                     int lsz, int scale) {
    __shared__ int sh_cnt;
    __shared__ int sh_tmp[MAX_GT];
    __shared__ int sh_pos[MAX_GT];
    __shared__ int sh_base[MAX_GT];
    __shared__ unsigned long long shKey[MAX_GT];
    __shared__ float shGT[MAX_GT][CCH];
    __shared__ float shRed[NT];

    const int tid = threadIdx.x;
    const int b = blockIdx.x;
    const int lsq = lsz * lsz;
    const int M = 3 * lsq;                       // rows per sample == #preds
    const size_t tbase = (size_t)b * (size_t)M * CCH;          // targets: [b][m][c]
    const size_t pbase = (size_t)b * 3 * CCH * (size_t)lsq;    // preds:   [b][ch][i][j]

    if (tid == 0) sh_cnt = 0;
    if (tid < MAX_GT) shKey[tid] = 0ull;
    __syncthreads();

    // ---- Phase A: scan conf channel, collect GT positions ----
    for (int m = tid; m < M; m += NT) {
        float conf = tgt[tbase + (size_t)m * CCH + 4];
        if (conf > 0.5f) {
            int s = atomicAdd(&sh_cnt, 1);
            if (s < MAX_GT) sh_tmp[s] = m;
        }
    }
    __syncthreads();
    int cnt = sh_cnt;
    if (cnt > MAX_GT) cnt = MAX_GT;
    // rank-sort by position -> deterministic order (also ascending indices for TDM gather)
    if (tid < cnt) {
        int mv = sh_tmp[tid];
        int r = 0;
        for (int k = 0; k < cnt; ++k) r += (sh_tmp[k] < mv) ? 1 : 0;
        sh_pos[r] = mv;
    }
    // pad unused GT slots with boxes that always give IoU == 0
    if (tid < MAX_GT && tid >= cnt) {
        shGT[tid][0] = -1e9f; shGT[tid][1] = -1e9f;
        shGT[tid][2] = 0.0f;  shGT[tid][3] = 0.0f;
    }
    __syncthreads();

    // ---- Phase A2a: TDM gather of first <=16 GT rows (85 f32 each) into LDS ----
    // D# per CDNA5 ISA: gather_mode, 16-bit indices,
    // data_size=4B, tensor_dim0=stride=tile_dim0=85, tensor_dim1=M, tile_dim1=#rows.
    if (tid < 32 && cnt > 0) {
        const int n1 = (cnt < 16) ? cnt : 16;
        unsigned ldsoff = (unsigned)(uintptr_t)(&shGT[0][0]);
        unsigned long long gaddr =
            (unsigned long long)(uintptr_t)tgt + tbase * sizeof(float);

        v4i G0, G2, G3;
        v8i G1;
        G0.x = rfl(0x80000001u);                                  // count=1 | gather_mode
        G0.y = rfl(ldsoff);                                       // lds_addr
        G0.z = rfl((unsigned)(gaddr & 0xFFFFFFFFull));            // global_addr lo
        G0.w = rfl((unsigned)((gaddr >> 32) & 0x01FFFFFFull) | (2u << 30)); // hi | type=2

        G1.s0 = rfl(2u << 16);                                    // data_size = 4 bytes
        G1.s1 = rfl(85u << 16);                                   // tensor_dim0 lo16
        G1.s2 = rfl(((unsigned)M & 0xFFFFu) << 16);               // tensor_dim1 lo16
        G1.s3 = rfl(85u << 16);                                   // tile_dim0 = 85
        G1.s4 = rfl((unsigned)n1);                                // tile_dim1 = #indices
        G1.s5 = rfl(85u);                                         // tensor_dim0_stride lo32
        G1.s6 = rfl(0u);
        G1.s7 = rfl(0u);

        unsigned id[16];
#pragma unroll
        for (int k = 0; k < 16; ++k) id[k] = (k < n1) ? (unsigned)sh_pos[k] : 0u;
        G2.x = rfl(id[0]  | (id[1]  << 16));
        G2.y = rfl(id[2]  | (id[3]  << 16));
        G2.z = rfl(id[4]  | (id[5]  << 16));
        G2.w = rfl(id[6]  | (id[7]  << 16));
        G3.x = rfl(id[8]  | (id[9]  << 16));
        G3.y = rfl(id[10] | (id[11] << 16));
        G3.z = rfl(id[12] | (id[13] << 16));
        G3.w = rfl(id[14] | (id[15] << 16));

        asm volatile("tensor_load_to_lds %0, %1, %2, %3"
                     :: "s"(G0), "s"(G1), "s"(G2), "s"(G3)
                     : "memory");
    }

    // ---- Phase A2b: async-stage remaining GT rows (16..cnt-1) into LDS ----
    for (int e = 16 * CCH + tid; e < cnt * CCH; e += NT) {
        int g = e / CCH;
        int c = e - g * CCH;
        unsigned goff = (unsigned)((tbase + (size_t)sh_pos[g] * CCH + (size_t)c) * sizeof(float));
        unsigned loff = (unsigned)(uintptr_t)(&shGT[g][c]);
        asm volatile("global_load_async_to_lds_b32 %0, %1, %2"
                     :: "v"(loff), "v"(goff), "s"(tgt)
                     : "memory");
    }
    __builtin_amdgcn_s_wait_tensorcnt(0);
    asm volatile("s_wait_asynccnt 0x0" ::: "memory");
    __syncthreads();

    const float anw0 = d_anch[scale][0][0], anh0 = d_anch[scale][0][1];
    const float anw1 = d_anch[scale][1][0], anh1 = d_anch[scale][1][1];
    const float anw2 = d_anch[scale][2][0], anh2 = d_anch[scale][2][1];
    const float flsz = (float)lsz;

    float acc = 0.0f;

    // ---- Phase B: per-pred decode, IoU vs 20 GTs, argmax keys, noobj BCE ----
    unsigned long long key[MAX_GT];
#pragma unroll
    for (int g = 0; g < MAX_GT; ++g) key[g] = 0ull;

    for (int j0 = tid; j0 < M; j0 += NT) {
        int a = j0 / lsq;
        int rem = j0 - a * lsq;
        int row = rem / lsz;
        int col = rem - row * lsz;
        size_t base = pbase + (size_t)(a * CCH) * lsq + (size_t)rem;

        if (j0 + NT < M) {
            int jn = j0 + NT;
            int an = jn / lsq;
            int remn = jn - an * lsq;
            __builtin_prefetch(&pred[pbase + (size_t)(an * CCH) * lsq + (size_t)remn], 0, 1);
        }

        float r0 = pred[base];
        float r1 = pred[base + (size_t)lsq];
        float r2 = pred[base + 2 * (size_t)lsq];
        float r3 = pred[base + 3 * (size_t)lsq];
        float r4 = pred[base + 4 * (size_t)lsq];

        float aw = (a == 0) ? anw0 : ((a == 1) ? anw1 : anw2);
        float ah = (a == 0) ? anh0 : ((a == 1) ? anh1 : anh2);
        float px = ((float)col + sigm(r0)) / flsz;
        float py = ((float)row + sigm(r1)) / flsz;
        float pw = aw * expf(r2);
        float ph = ah * expf(r3);
        float s4 = sigm(r4);

        float pmax = 0.0f;
#pragma unroll
        for (int g = 0; g < MAX_GT; ++g) {
            float iou = iou_one(px, py, pw, ph,
                                shGT[g][0], shGT[g][1], shGT[g][2], shGT[g][3]);
            pmax = fmaxf(pmax, iou);
            unsigned long long k =
                ((unsigned long long)__float_as_uint(iou) << 32) |
                (unsigned long long)(0xFFFFFFFFu - (unsigned)j0);
            key[g] = (k > key[g]) ? k : key[g];
        }
        if (pmax < 0.5f) {
            float p = clipp(s4);
            acc += -logf(1.0f - p);   // bce(s4, 0)
        }
    }
#pragma unroll
    for (int g = 0; g < MAX_GT; ++g) atomicMax(&shKey[g], key[g]);
    __syncthreads();

    // ---- Phase D: selected-pred terms (reg + positive conf + noobj dedup) ----
    if (tid < cnt) {
        int g = tid;
        unsigned bidx = 0xFFFFFFFFu - (unsigned)(shKey[g] & 0xFFFFFFFFull);
        int a = (int)bidx / lsq;
        int rem = (int)bidx - a * lsq;
        int row = rem / lsz;
        int col = rem - row * lsz;
        int sbase = a * CCH * lsq + rem;
        sh_base[g] = sbase;

        size_t base = pbase + (size_t)sbase;
        float r0 = pred[base];
        float r1 = pred[base + (size_t)lsq];
        float r2 = pred[base + 2 * (size_t)lsq];
        float r3 = pred[base + 3 * (size_t)lsq];
        float r4 = pred[base + 4 * (size_t)lsq];
        float aw = (a == 0) ? anw0 : ((a == 1) ? anw1 : anw2);
        float ah = (a == 0) ? anh0 : ((a == 1) ? anh1 : anh2);
        float px = ((float)col + sigm(r0)) / flsz;
        float py = ((float)row + sigm(r1)) / flsz;
        float pw = aw * expf(r2);
        float ph = ah * expf(r3);
        float s4 = sigm(r4);

        float gx = shGT[g][0], gy = shGT[g][1], gw = shGT[g][2], gh = shGT[g][3];
        float dx = px - gx, dy = py - gy, dw = pw - gw, dh = ph - gh;
        float reg = (dx * dx + dy * dy + dw * dw + dh * dh) * (2.0f - gw * gh);
        float p4 = clipp(s4);
        float v = reg - logf(p4);                 // reg + bce(s4, 1)

        // distinct-best-index dedup for the noobj scatter-multiply
        bool first = true;
        for (int gp = 0; gp < g; ++gp) {
            unsigned bp = 0xFFFFFFFFu - (unsigned)(shKey[gp] & 0xFFFFFFFFull);
            if (bp == bidx) first = false;
        }
        if (first) {
            float pmax = 0.0f;
#pragma unroll
            for (int g2 = 0; g2 < MAX_GT; ++g2) {
                pmax = fmaxf(pmax, iou_one(px, py, pw, ph,
                                           shGT[g2][0], shGT[g2][1],
                                           shGT[g2][2], shGT[g2][3]));
            }
            if (pmax < 0.5f) v += logf(1.0f - p4);  // subtract bce(s4, 0)
        }
        acc += v;
    }
    __syncthreads();

    // ---- Class BCE for selected preds: cnt * 80 terms across the block ----
    for (int e = tid; e < cnt * NUM_CLASSES; e += NT) {
        int g = e / NUM_CLASSES;
        int cc = e - g * NUM_CLASSES;
        float rr = pred[pbase + (size_t)sh_base[g] + (size_t)(5 + cc) * lsq];
        float p = clipp(sigm(rr));
        float t = shGT[g][5 + cc];
        acc += -t * logf(p) - (1.0f - t) * logf(1.0f - p);
    }

    // ---- Deterministic block tree reduction ----
    shRed[tid] = acc;
    __syncthreads();
    for (int s = NT / 2; s > 0; s >>= 1) {
        if (tid < s) shRed[tid] += shRed[tid + s];
        __syncthreads();
    }
    if (tid == 0) partial[b] = shRed[0];
}

__global__ __launch_bounds__(256)
void yolo_finalize(const float* __restrict__ partial, float* __restrict__ out, int n) {
    __shared__ float red[256];
    int tid = threadIdx.x;
    red[tid] = (tid < n) ? partial[tid] : 0.0f;
    __syncthreads();
    for (int s = 128; s > 0; s >>= 1) {
        if (tid < s) red[tid] += red[tid + s];
        __syncthreads();
    }
    if (tid == 0) out[0] = red[0];
}

extern "C" void kernel_launch(void* const* d_in, const int* in_sizes, int n_in,
                              void* d_out, int out_size, void* d_ws, size_t ws_size,
                              hipStream_t stream) {
    // setup_inputs() dict order: pred0, targets0, pred1, targets1, pred2, targets2
    const float* pred0 = (const float*)d_in[0];
    const float* tgt0  = (const float*)d_in[1];
    const float* pred1 = (const float*)d_in[2];
    const float* tgt1  = (const float*)d_in[3];
    const float* pred2 = (const float*)d_in[4];
    const float* tgt2  = (const float*)d_in[5];
    float* ws = (float*)d_ws;

    yolo_scale_loss<256><<<64, 256, 0, stream>>>(pred0, tgt0, ws + 0,   13, 0);
    yolo_scale_loss<512><<<64, 512, 0, stream>>>(pred1, tgt1, ws + 64,  26, 1);
    yolo_scale_loss<512><<<64, 512, 0, stream>>>(pred2, tgt2, ws + 128, 52, 2);
    yolo_finalize<<<1, 256, 0, stream>>>(ws, (float*)d_out, 192);
}